// HierNet_88510686036544
// MI455X (gfx1250) — compile-verified
//
#include <hip/hip_runtime.h>
#include <math.h>

#define N_NODES 30000
#define N_EDGES 480000
#define BATCH_B 64
#define CH      128
#define LAYERS  3
#define HLSZ    64
#define FOURC   512

typedef __attribute__((ext_vector_type(2))) float v2f;
typedef __attribute__((ext_vector_type(8))) float v8f;

// ---------------------------------------------------------------- utilities

__device__ __forceinline__ float warp_sum(float v) {
#pragma unroll
  for (int off = 16; off > 0; off >>= 1) v += __shfl_xor(v, off, 32);
  return v;
}

// float atomic-max via signed/unsigned bit ordering trick (init to -inf)
__device__ __forceinline__ void atomicMaxFloat(float* addr, float value) {
  if (value >= 0.0f) atomicMax((int*)addr, __float_as_int(value));
  else               atomicMin((unsigned int*)addr, __float_as_uint(value));
}

__global__ void fill_kernel(float* __restrict__ p, float v, size_t n) {
  size_t i = (size_t)blockIdx.x * blockDim.x + threadIdx.x;
  if (i < n) p[i] = v;
}

__global__ void add_bias_kernel(const float* __restrict__ a, const float* __restrict__ b,
                                float* __restrict__ o, int n) {
  int i = blockIdx.x * blockDim.x + threadIdx.x;
  if (i < n) o[i] = a[i] + b[i];
}

// ------------------------------------------------- exact-f32 WMMA GEMM core
// Y[M x Nout] = relu?( X[M x K] @ W (+ bias) (+ addend) )
// wTrans==0: W is [K x Nout] row-major (Y = X@W)
// wTrans==1: W is [Nout x K] row-major (Y = X@W.T)
// One wave computes a 16x32 output block (two 16x16 WMMA tiles, two
// independent accumulator chains) via V_WMMA_F32_16X16X4_F32 (exact fp32).
// A fragments are loaded once and reused by both tiles.
// PRECONDITIONS (hold for every call in this network): M % 16 == 0,
// Nout % 32 == 0, K % 4 == 0 -> no ragged tiles, no per-lane predication,
// so EXEC stays all-ones through the WMMA loop (ISA requirement).
__global__ void gemm_wmma_f32(const float* __restrict__ X,
                              const float* __restrict__ W,
                              const float* __restrict__ bias,
                              const float* __restrict__ addend,
                              float* __restrict__ Y,
                              int M, int K, int Nout,
                              int wTrans, int doRelu)
{
  const int lane = threadIdx.x;                               // 0..31
  const int m0   = blockIdx.x * 16;
  const int n0   = (blockIdx.y * blockDim.y + threadIdx.y) * 32;
  if (n0 >= Nout) return;                                     // uniform per wave
  (void)M;

  const int mrow  = m0 + (lane & 15);
  const int ncol  = n0 + (lane & 15);
  const int khalf = (lane >> 4) << 1;                         // 0 or 2

  const float* __restrict__ arow = X + (size_t)mrow * K + khalf;
  v8f acc0 = {};
  v8f acc1 = {};

  if (wTrans) {
    const float* __restrict__ bcol0 = W + (size_t)ncol * K + khalf;
    const float* __restrict__ bcol1 = bcol0 + (size_t)16 * K;
    __builtin_prefetch(arow, 0, 0);
    __builtin_prefetch(bcol0, 0, 0);
    __builtin_prefetch(bcol1, 0, 0);
#pragma unroll 4
    for (int k = 0; k < K; k += 4) {
      v2f a  = *(const v2f*)(arow + k);                       // global_load_b64
      v2f b0 = *(const v2f*)(bcol0 + k);                      // global_load_b64
      v2f b1 = *(const v2f*)(bcol1 + k);                      // global_load_b64
      acc0 = __builtin_amdgcn_wmma_f32_16x16x4_f32(false, a, false, b0,
                                                   (short)0, acc0, false, false);
      acc1 = __builtin_amdgcn_wmma_f32_16x16x4_f32(false, a, false, b1,
                                                   (short)0, acc1, false, false);
    }
  } else {
    const float* __restrict__ bptr = W + (size_t)khalf * Nout + ncol;
    __builtin_prefetch(arow, 0, 0);
    __builtin_prefetch(bptr, 0, 0);
#pragma unroll 4
    for (int k = 0; k < K; k += 4) {
      v2f a = *(const v2f*)(arow + k);                        // global_load_b64
      v2f b0, b1;
      b0.x = bptr[(size_t)k * Nout];
      b0.y = bptr[(size_t)(k + 1) * Nout];
      b1.x = bptr[(size_t)k * Nout + 16];
      b1.y = bptr[(size_t)(k + 1) * Nout + 16];
      acc0 = __builtin_amdgcn_wmma_f32_16x16x4_f32(false, a, false, b0,
                                                   (short)0, acc0, false, false);
      acc1 = __builtin_amdgcn_wmma_f32_16x16x4_f32(false, a, false, b1,
                                                   (short)0, acc1, false, false);
    }
  }

  const float bval0 = bias ? bias[ncol]      : 0.0f;
  const float bval1 = bias ? bias[ncol + 16] : 0.0f;
  const int   rbase = (lane >> 4) * 8;                        // C/D: M = r (+8 for hi lanes)
#pragma unroll
  for (int r = 0; r < 8; ++r) {
    size_t off0 = (size_t)(m0 + rbase + r) * Nout + ncol;
    float v0 = acc0[r] + bval0;
    float v1 = acc1[r] + bval1;
    if (addend) { v0 += addend[off0]; v1 += addend[off0 + 16]; }
    if (doRelu) { v0 = fmaxf(v0, 0.0f); v1 = fmaxf(v1, 0.0f); }
    Y[off0]      = v0;
    Y[off0 + 16] = v1;
  }
}

// ------------------------------------------------------ edge attention path

// wave per edge: logits[e] = dot(q[dst], k[src]) * scale ; segment max over dst
__global__ void edge_logits_max_kernel(const float* __restrict__ q,
                                       const float* __restrict__ k,
                                       const int* __restrict__ src,
                                       const int* __restrict__ dst,
                                       float* __restrict__ logits,
                                       float* __restrict__ segmax,
                                       int E, float scale)
{
  int e = blockIdx.x * blockDim.y + threadIdx.y;
  if (e >= E) return;
  int lane = threadIdx.x;
  int s = src[e], d = dst[e];
  float4 qv = ((const float4*)(q + (size_t)d * CH))[lane];
  float4 kv = ((const float4*)(k + (size_t)s * CH))[lane];
  float p = qv.x * kv.x + qv.y * kv.y + qv.z * kv.z + qv.w * kv.w;
  p = warp_sum(p) * scale;
  if (lane == 0) {
    logits[e] = p;
    atomicMaxFloat(&segmax[d], p);
  }
}

__global__ void edge_exp_kernel(const int* __restrict__ dst, float* __restrict__ logits,
                                const float* __restrict__ segmax, float* __restrict__ denom,
                                int E)
{
  int e = blockIdx.x * blockDim.x + threadIdx.x;
  if (e >= E) return;
  int d = dst[e];
  float ex = __expf(logits[e] - segmax[d]);
  logits[e] = ex;
  atomicAdd(&denom[d], ex);
}

// wave per edge: agg[dst] += (e/denom[dst]) * v[src]
__global__ void edge_agg_kernel(const int* __restrict__ src, const int* __restrict__ dst,
                                const float* __restrict__ logits, const float* __restrict__ denom,
                                const float* __restrict__ v, float* __restrict__ agg, int E)
{
  int e = blockIdx.x * blockDim.y + threadIdx.y;
  if (e >= E) return;
  int lane = threadIdx.x;
  int s = src[e], d = dst[e];
  float w = logits[e] / denom[d];
  float4 vv = ((const float4*)(v + (size_t)s * CH))[lane];
  float* dr = agg + (size_t)d * CH + lane * 4;
  atomicAdd(dr + 0, w * vv.x);
  atomicAdd(dr + 1, w * vv.y);
  atomicAdd(dr + 2, w * vv.z);
  atomicAdd(dr + 3, w * vv.w);
}

// ---------------------------------------------------------------- LSTM path

// z layout [N, 4C] = [i | f | g | o]; updates h,c in place ([N, C])
__global__ void lstm_gates_kernel(const float* __restrict__ z, float* __restrict__ h,
                                  float* __restrict__ c, int total)
{
  int idx = blockIdx.x * blockDim.x + threadIdx.x;
  if (idx >= total) return;
  int n  = idx >> 7;          // /CH
  int ch = idx & (CH - 1);
  const float* zr = z + (size_t)n * FOURC;
  float zi = zr[ch], zf = zr[CH + ch], zg = zr[2 * CH + ch], zo = zr[3 * CH + ch];
  float ig = 1.0f / (1.0f + __expf(-zi));
  float fg = 1.0f / (1.0f + __expf(-zf));
  float og = 1.0f / (1.0f + __expf(-zo));
  float cn = fg * c[idx] + ig * tanhf(zg);
  c[idx] = cn;
  h[idx] = og * tanhf(cn);
}

// wave per node: s[n] += dot(h[n,:], attWpart[0:128])
__global__ void attn_score_kernel(const float* __restrict__ h, const float* __restrict__ attW,
                                  float* __restrict__ s)
{
  int n = blockIdx.x * blockDim.y + threadIdx.y;
  if (n >= N_NODES) return;
  int lane = threadIdx.x;
  float4 hv = ((const float4*)(h + (size_t)n * CH))[lane];
  float4 wv = ((const float4*)attW)[lane];
  float p = hv.x * wv.x + hv.y * wv.y + hv.z * wv.z + hv.w * wv.w;
  p = warp_sum(p);
  if (lane == 0) s[n] += p;
}

// wave per node: softmax over 3 layer scores, weighted sum of xs -> xout
__global__ void jk_combine_kernel(const float* __restrict__ xs, const float* __restrict__ s,
                                  float* __restrict__ xout)
{
  int n = blockIdx.x * blockDim.y + threadIdx.y;
  if (n >= N_NODES) return;
  int lane = threadIdx.x;
  float s0 = s[n], s1 = s[N_NODES + n], s2 = s[2 * N_NODES + n];
  float m  = fmaxf(s0, fmaxf(s1, s2));
  float e0 = __expf(s0 - m), e1 = __expf(s1 - m), e2 = __expf(s2 - m);
  float inv = 1.0f / (e0 + e1 + e2);
  float a0 = e0 * inv, a1 = e1 * inv, a2 = e2 * inv;
  size_t off = (size_t)n * CH + lane * 4;
  float4 x0 = *(const float4*)(xs + off);
  float4 x1 = *(const float4*)(xs + (size_t)N_NODES * CH + off);
  float4 x2 = *(const float4*)(xs + 2 * (size_t)N_NODES * CH + off);
  float4 r;
  r.x = a0 * x0.x + a1 * x1.x + a2 * x2.x;
  r.y = a0 * x0.y + a1 * x1.y + a2 * x2.y;
  r.z = a0 * x0.z + a1 * x1.z + a2 * x2.z;
  r.w = a0 * x0.w + a1 * x1.w + a2 * x2.w;
  *(float4*)(xout + off) = r;
}

// ------------------------------------------------------------- gate pooling

__global__ void gate_logit_kernel(const float* __restrict__ ghid, const float* __restrict__ W2,
                                  const float* __restrict__ b2, float* __restrict__ glog)
{
  int n = blockIdx.x * blockDim.y + threadIdx.y;
  if (n >= N_NODES) return;
  int lane = threadIdx.x;
  float4 hv = ((const float4*)(ghid + (size_t)n * CH))[lane];
  float4 wv = ((const float4*)W2)[lane];
  float p = hv.x * wv.x + hv.y * wv.y + hv.z * wv.z + hv.w * wv.w;
  p = warp_sum(p);
  if (lane == 0) glog[n] = p + b2[0];
}

__global__ void seg_max_kernel(const float* __restrict__ glog, const int* __restrict__ batch,
                               float* __restrict__ gmax, int n)
{
  int i = blockIdx.x * blockDim.x + threadIdx.x;
  if (i < n) atomicMaxFloat(&gmax[batch[i]], glog[i]);
}

__global__ void seg_exp_kernel(float* __restrict__ glog, const int* __restrict__ batch,
                               const float* __restrict__ gmax, float* __restrict__ gden, int n)
{
  int i = blockIdx.x * blockDim.x + threadIdx.x;
  if (i >= n) return;
  int b = batch[i];
  float ex = __expf(glog[i] - gmax[b]);
  glog[i] = ex;
  atomicAdd(&gden[b], ex);
}

// wave per node: pooled[batch[n]] += (e/den) * x[n]
__global__ void pool_accum_kernel(const float* __restrict__ x, const float* __restrict__ ge,
                                  const float* __restrict__ gden, const int* __restrict__ batch,
                                  float* __restrict__ pooled)
{
  int n = blockIdx.x * blockDim.y + threadIdx.y;
  if (n >= N_NODES) return;
  int lane = threadIdx.x;
  int b = batch[n];
  float w = ge[n] / gden[b];
  float4 xv = ((const float4*)(x + (size_t)n * CH))[lane];
  float* dr = pooled + (size_t)b * CH + lane * 4;
  atomicAdd(dr + 0, w * xv.x);
  atomicAdd(dr + 1, w * xv.y);
  atomicAdd(dr + 2, w * xv.z);
  atomicAdd(dr + 3, w * xv.w);
}

// --------------------------------------------------------------- final MLP

__global__ void mlp_kernel(const float* __restrict__ poolP, const float* __restrict__ poolT,
                           const float* __restrict__ hls,
                           const float* __restrict__ W0, const float* __restrict__ b0,
                           const float* __restrict__ W1, const float* __restrict__ b1,
                           const float* __restrict__ W2, const float* __restrict__ b2,
                           float* __restrict__ out)
{
  int b = threadIdx.x;
  if (b >= BATCH_B) return;
  float h0[64];
  for (int j = 0; j < 64; ++j) {
    float acc = b0[j];
    for (int k2 = 0; k2 < CH;   ++k2) acc += poolP[b * CH + k2]  * W0[k2 * 64 + j];
    for (int k2 = 0; k2 < CH;   ++k2) acc += poolT[b * CH + k2]  * W0[(CH + k2) * 64 + j];
    for (int k2 = 0; k2 < HLSZ; ++k2) acc += hls[b * HLSZ + k2]  * W0[(2 * CH + k2) * 64 + j];
    h0[j] = fmaxf(acc, 0.0f);
  }
  float h1[64];
  for (int j = 0; j < 64; ++j) {
    float acc = b1[j];
    for (int k2 = 0; k2 < 64; ++k2) acc += h0[k2] * W1[k2 * 64 + j];
    h1[j] = fmaxf(acc, 0.0f);
  }
  float acc = b2[0];
  for (int k2 = 0; k2 < 64; ++k2) acc += h1[k2] * W2[k2];
  out[b] = acc;
}

// ------------------------------------------------------------------- driver

extern "C" void kernel_launch(void* const* d_in, const int* in_sizes, int n_in,
                              void* d_out, int out_size, void* d_ws, size_t ws_size,
                              hipStream_t stream)
{
  (void)in_sizes; (void)n_in; (void)out_size; (void)ws_size;

  const float* x        = (const float*)d_in[0];
  const int*   eidx     = (const int*)  d_in[1];
  const int*   batch    = (const int*)  d_in[2];
  const float* hls_attr = (const float*)d_in[3];
  const float* Wq    = (const float*)d_in[4];
  const float* bq    = (const float*)d_in[5];
  const float* Wk    = (const float*)d_in[6];
  const float* bk    = (const float*)d_in[7];
  const float* Wv    = (const float*)d_in[8];
  const float* bv    = (const float*)d_in[9];
  const float* Wskip = (const float*)d_in[10];
  const float* bskip = (const float*)d_in[11];
  const float* Wih_f = (const float*)d_in[12];
  const float* Whh_f = (const float*)d_in[13];
  const float* bih_f = (const float*)d_in[14];
  const float* bhh_f = (const float*)d_in[15];
  const float* Wih_b = (const float*)d_in[16];
  const float* Whh_b = (const float*)d_in[17];
  const float* bih_b = (const float*)d_in[18];
  const float* bhh_b = (const float*)d_in[19];
  const float* attW  = (const float*)d_in[20];
  // d_in[21] (att_b): uniform shift across layer axis -> softmax-invariant, omitted
  const float* gPW1 = (const float*)d_in[22];
  const float* gPb1 = (const float*)d_in[23];
  const float* gPW2 = (const float*)d_in[24];
  const float* gPb2 = (const float*)d_in[25];
  const float* gTW1 = (const float*)d_in[26];
  const float* gTb1 = (const float*)d_in[27];
  const float* gTW2 = (const float*)d_in[28];
  const float* gTb2 = (const float*)d_in[29];
  const float* mW0  = (const float*)d_in[30];
  const float* mb0  = (const float*)d_in[31];
  const float* mW1  = (const float*)d_in[32];
  const float* mb1  = (const float*)d_in[33];
  const float* mW2  = (const float*)d_in[34];
  const float* mb2  = (const float*)d_in[35];

  const int* src = eidx;            // edge_index[0]
  const int* dst = eidx + N_EDGES;  // edge_index[1]

  const size_t NC = (size_t)N_NODES * CH;
  float* ws    = (float*)d_ws;
  float* xs    = ws;                          // 3*NC  (layer outputs)
  float* zbuf  = xs + 3 * NC;                 // 4*NC  (q|k|v|agg, later LSTM z, later gate hidden)
  float* hbuf  = zbuf + 4 * NC;               // NC
  float* cbuf  = hbuf + NC;                   // NC
  float* xjk   = cbuf + NC;                   // NC
  float* sarr  = xjk + NC;                    // 3*N
  float* elog  = sarr + 3 * N_NODES;          // E
  float* smax  = elog + N_EDGES;              // N
  float* sden  = smax + N_NODES;              // N
  float* glog  = sden + N_NODES;              // N
  float* gmax  = glog + N_NODES;              // B
  float* gden  = gmax + BATCH_B;              // B
  float* poolP = gden + BATCH_B;              // B*CH
  float* poolT = poolP + BATCH_B * CH;        // B*CH
  float* bcomb = poolT + BATCH_B * CH;        // 4C

  float* qb  = zbuf;
  float* kb  = zbuf + NC;
  float* vb  = zbuf + 2 * NC;
  float* agg = zbuf + 3 * NC;

  const float scale = 1.0f / sqrtf((float)CH);

  auto gemm = [&](const float* X, const float* W, const float* bias, const float* add,
                  float* Y, int M, int K, int Nout, int trans, int relu) {
    dim3 blk(32, 4);
    dim3 grd((unsigned)((M + 15) / 16), (unsigned)((Nout / 32 + 3) / 4));
    gemm_wmma_f32<<<grd, blk, 0, stream>>>(X, W, bias, add, Y, M, K, Nout, trans, relu);
  };
  auto fill = [&](float* p, float v, size_t n) {
    fill_kernel<<<dim3((unsigned)((n + 255) / 256)), dim3(256), 0, stream>>>(p, v, n);
  };

  const dim3 nodeGrd((N_NODES + 7) / 8), nodeBlk(32, 8);
  const dim3 edgeGrd((N_EDGES + 7) / 8), edgeBlk(32, 8);

  // ---- 3 TransformerConv layers ----
  const float* xcur = x;
  for (int i = 0; i < LAYERS; ++i) {
    const size_t wo = (size_t)i * CH * CH;
    gemm(xcur, Wq + wo, bq + i * CH, nullptr, qb, N_NODES, CH, CH, 0, 0);
    gemm(xcur, Wk + wo, bk + i * CH, nullptr, kb, N_NODES, CH, CH, 0, 0);
    gemm(xcur, Wv + wo, bv + i * CH, nullptr, vb, N_NODES, CH, CH, 0, 0);

    fill(smax, -INFINITY, N_NODES);
    fill(sden, 0.0f, N_NODES);
    fill(agg, 0.0f, NC);
    edge_logits_max_kernel<<<edgeGrd, edgeBlk, 0, stream>>>(qb, kb, src, dst, elog, smax,
                                                            N_EDGES, scale);
    edge_exp_kernel<<<(N_EDGES + 255) / 256, 256, 0, stream>>>(dst, elog, smax, sden, N_EDGES);
    edge_agg_kernel<<<edgeGrd, edgeBlk, 0, stream>>>(src, dst, elog, sden, vb, agg, N_EDGES);

    float* xnext = xs + (size_t)i * NC;
    gemm(xcur, Wskip + wo, bskip + i * CH, agg, xnext, N_NODES, CH, CH, 0, 1);
    xcur = xnext;
  }

  // ---- BiLSTM over the 3 layer outputs + per-layer attention scores ----
  fill(sarr, 0.0f, 3 * N_NODES);

  // forward direction
  add_bias_kernel<<<2, 256, 0, stream>>>(bih_f, bhh_f, bcomb, FOURC);
  fill(hbuf, 0.0f, NC);
  fill(cbuf, 0.0f, NC);
  for (int t = 0; t < LAYERS; ++t) {
    gemm(xs + (size_t)t * NC, Wih_f, bcomb, nullptr, zbuf, N_NODES, CH, FOURC, 1, 0);
    if (t > 0) gemm(hbuf, Whh_f, nullptr, zbuf, zbuf, N_NODES, CH, FOURC, 1, 0);
    lstm_gates_kernel<<<(unsigned)((NC + 255) / 256), 256, 0, stream>>>(zbuf, hbuf, cbuf, (int)NC);
    attn_score_kernel<<<nodeGrd, nodeBlk, 0, stream>>>(hbuf, attW, sarr + (size_t)t * N_NODES);
  }
  // backward direction
  add_bias_kernel<<<2, 256, 0, stream>>>(bih_b, bhh_b, bcomb, FOURC);
  fill(hbuf, 0.0f, NC);
  fill(cbuf, 0.0f, NC);
  for (int sidx = 0; sidx < LAYERS; ++sidx) {
    int t = LAYERS - 1 - sidx;
    gemm(xs + (size_t)t * NC, Wih_b, bcomb, nullptr, zbuf, N_NODES, CH, FOURC, 1, 0);
    if (sidx > 0) gemm(hbuf, Whh_b, nullptr, zbuf, zbuf, N_NODES, CH, FOURC, 1, 0);
    lstm_gates_kernel<<<(unsigned)((NC + 255) / 256), 256, 0, stream>>>(zbuf, hbuf, cbuf, (int)NC);
    attn_score_kernel<<<nodeGrd, nodeBlk, 0, stream>>>(hbuf, attW + CH, sarr + (size_t)t * N_NODES);
  }

  // ---- JK combine (softmax over layers) ----
  jk_combine_kernel<<<nodeGrd, nodeBlk, 0, stream>>>(xs, sarr, xjk);

  // ---- two gated pools ----
  auto gate_pool = [&](const float* W1, const float* b1, const float* W2, const float* b2,
                       float* pooled) {
    gemm(xjk, W1, b1, nullptr, zbuf, N_NODES, CH, CH, 0, 1);
    gate_logit_kernel<<<nodeGrd, nodeBlk, 0, stream>>>(zbuf, W2, b2, glog);
    fill(gmax, -INFINITY, BATCH_B);
    fill(gden, 0.0f, BATCH_B);
    seg_max_kernel<<<(N_NODES + 255) / 256, 256, 0, stream>>>(glog, batch, gmax, N_NODES);
    seg_exp_kernel<<<(N_NODES + 255) / 256, 256, 0, stream>>>(glog, batch, gmax, gden, N_NODES);
    fill(pooled, 0.0f, (size_t)BATCH_B * CH);
    pool_accum_kernel<<<nodeGrd, nodeBlk, 0, stream>>>(xjk, glog, gden, batch, pooled);
  };
  gate_pool(gPW1, gPb1, gPW2, gPb2, poolP);
  gate_pool(gTW1, gTb1, gTW2, gTb2, poolT);

  // ---- final MLP on B=64 graphs ----
  mlp_kernel<<<1, 64, 0, stream>>>(poolP, poolT, hls_attr, mW0, mb0, mW1, mb1, mW2, mb2,
                                   (float*)d_out);
}